// AngularLoss_60696477827573
// MI455X (gfx1250) — compile-verified
//
#include <hip/hip_runtime.h>
#include <hip/hip_bf16.h>

typedef float v2f __attribute__((ext_vector_type(2)));
typedef float v8f __attribute__((ext_vector_type(8)));

#define D 128           // feature dim (fixed by reference)
#define ROWS_PER_BLOCK 128
#define NBLOCKS_MAX 4096

// Pass 1: each wave computes f for a 16-row tile via WMMA diag trick.
// Block emits (max_b, sum_b = sum exp(f - max_b)) for its 128 rows.
__global__ __launch_bounds__(256) void angular_pass1(
    const float* __restrict__ an, const float* __restrict__ po,
    const float* __restrict__ ne, const int* __restrict__ alphaPtr,
    float* __restrict__ blockMax, float* __restrict__ blockSum) {
  const int tid  = threadIdx.x;
  const int lane = tid & 31;
  const int wave = tid >> 5;

  // Tile of 16 rows handled by this wave.
  const long long rowBase = (long long)blockIdx.x * ROWS_PER_BLOCK + (long long)wave * 16;
  const int  mrow  = lane & 15;          // row within tile (A: M across lanes; B: N across lanes)
  const int  khalf = (lane >> 4) << 1;   // K sub-offset: lanes 0-15 -> k+0..1, lanes 16-31 -> k+2..3

  const float* pa = an + (rowBase + mrow) * D + khalf;
  const float* pp = po + (rowBase + mrow) * D + khalf;
  const float* pn = ne + (rowBase + mrow) * D + khalf;

  v8f c1 = {}; // accumulates (a+p) x n^T  -> diag = apn
  v8f c2 = {}; // accumulates  a    x p^T  -> diag = a.p

#pragma unroll
  for (int k0 = 0; k0 < D; k0 += 4) {
    v2f a = *(const v2f*)(pa + k0);
    v2f p = *(const v2f*)(pp + k0);
    v2f n = *(const v2f*)(pn + k0);
    v2f s = a + p;
    // D = A(16x4) * B(4x16) + C ; A/B per-lane layouts coincide for this load pattern.
    c1 = __builtin_amdgcn_wmma_f32_16x16x4_f32(false, s, false, n, (short)0, c1, false, false);
    c2 = __builtin_amdgcn_wmma_f32_16x16x4_f32(false, a, false, p, (short)0, c2, false, false);
  }

  // Diagonal extraction: row m lives in VGPR (m&7) at lane m (m<8) or lane m+16 (m>=8).
  const int srcLane = (lane < 8) ? lane : (lane + 16);  // only meaningful for lanes 0..15
  float apn = 0.0f, app = 0.0f;
#pragma unroll
  for (int v = 0; v < 8; ++v) {
    float t1 = __shfl(c1[v], srcLane, 32);
    float t2 = __shfl(c2[v], srcLane, 32);
    if ((lane & 7) == v) { apn = t1; app = t2; }
  }

  // f = 4*t2*apn - 2*(1+t2)*ap  (t2 = tan(alpha_deg * 2pi/360)^2, alpha read on-device)
  const int   alpha = *alphaPtr;
  const float arad  = 6.28318530717958647692f * (float)alpha * (1.0f / 360.0f);
  const float tt    = tanf(arad);
  const float t2c   = tt * tt;
  const bool  valid = (lane < 16);
  const float f = valid ? (4.0f * t2c * apn - 2.0f * (1.0f + t2c) * app)
                        : -__builtin_inff();

  // Deterministic block reduction: max, then sum of exp(f - max).
  __shared__ float sm[256];
  sm[tid] = f;
  __syncthreads();
#pragma unroll
  for (int s = 128; s > 0; s >>= 1) {
    if (tid < s) sm[tid] = fmaxf(sm[tid], sm[tid + s]);
    __syncthreads();
  }
  const float bmax = sm[0];
  __syncthreads();

  sm[tid] = valid ? __expf(f - bmax) : 0.0f;
  __syncthreads();
#pragma unroll
  for (int s = 128; s > 0; s >>= 1) {
    if (tid < s) sm[tid] = sm[tid] + sm[tid + s];
    __syncthreads();
  }
  if (tid == 0) {
    blockMax[blockIdx.x] = bmax;
    blockSum[blockIdx.x] = sm[0];
  }
}

// Pass 2: single block combines (max_b, sum_b) pairs deterministically.
__global__ __launch_bounds__(256) void angular_pass2(
    const float* __restrict__ blockMax, const float* __restrict__ blockSum,
    float* __restrict__ out, int nblocks) {
  const int tid = threadIdx.x;
  float m = -__builtin_inff();
  float s = 0.0f;
  for (int i = tid; i < nblocks; i += 256) {
    const float bm = blockMax[i];
    const float bs = blockSum[i];
    const float nm = fmaxf(m, bm);
    s = s * __expf(m - nm) + bs * __expf(bm - nm);
    m = nm;
  }
  __shared__ float smx[256];
  __shared__ float ssm[256];
  smx[tid] = m; ssm[tid] = s;
  __syncthreads();
#pragma unroll
  for (int st = 128; st > 0; st >>= 1) {
    if (tid < st) {
      const float m2 = smx[tid + st];
      const float s2 = ssm[tid + st];
      const float nm = fmaxf(smx[tid], m2);
      ssm[tid] = ssm[tid] * __expf(smx[tid] - nm) + s2 * __expf(m2 - nm);
      smx[tid] = nm;
    }
    __syncthreads();
  }
  if (tid == 0) out[0] = logf(ssm[0]) + smx[0];
}

extern "C" void kernel_launch(void* const* d_in, const int* in_sizes, int n_in,
                              void* d_out, int out_size, void* d_ws, size_t ws_size,
                              hipStream_t stream) {
  const float* an    = (const float*)d_in[0];
  const float* po    = (const float*)d_in[1];
  const float* ne    = (const float*)d_in[2];
  const int*   alpha = (const int*)d_in[3];

  const int rows    = in_sizes[0] / D;          // 262144
  const int nblocks = rows / ROWS_PER_BLOCK;    // 2048 (exact for reference shape)

  float* bMax = (float*)d_ws;
  float* bSum = bMax + NBLOCKS_MAX;

  angular_pass1<<<nblocks, 256, 0, stream>>>(an, po, ne, alpha, bMax, bSum);
  angular_pass2<<<1, 256, 0, stream>>>(bMax, bSum, (float*)d_out, nblocks);
}